// OR_MLP_6167573037692
// MI455X (gfx1250) — compile-verified
//
#include <hip/hip_runtime.h>

// ---------------------------------------------------------------------------
// Autoregressive tiny-MLP rollout on CDNA5 (gfx1250).
//
// Strategy: 16 batch elements per wave32. Layer 1 (x[20] @ W1[20x6]) is done
// as H^T = W1^T * X^T with V_WMMA_F32_16X16X4_F32 (5 K-chunks of 4, K=20
// exactly). W1^T is duplicated into A rows 0-5 AND 8-13 so that, in the C/D
// layout (M = vgpr + 8*(lane>=16), N = lane%16), *every* lane ends up with
// h0..h5 of batch (lane%16) in acc vgprs 0..5. tanh, the 6x2 second layer,
// and the sliding prediction window are then purely per-lane VALU — zero
// cross-lane traffic in the 251-step recurrence.
//
// v2: tanh uses raw v_rcp_f32 (__builtin_amdgcn_rcpf) instead of the IEEE
// division macro — the previous build spent ~12 dependent VALU ops per hidden
// unit in v_div_scale/v_div_fmas/v_div_fixup, dominating the latency-bound
// inner loop. Now tanh = v_exp_f32 + v_add + v_rcp_f32 + v_fma.
// ---------------------------------------------------------------------------

typedef __attribute__((ext_vector_type(2))) float v2f;
typedef __attribute__((ext_vector_type(8))) float v8f;

#define WS    5
#define OS    2
#define HID   6
#define DT    4      // D = 2*OS
#define T_IN  256
#define T_OUT 251    // 5 warm-up + 246 scan steps

__device__ __forceinline__ float4 ld_row(const float* __restrict__ traj, int b, int t) {
    return *reinterpret_cast<const float4*>(traj + ((size_t)b * T_IN + t) * DT);
}

__device__ __forceinline__ float tanh_fast(float v) {
    // tanh(x) = 1 - 2/(exp(2x)+1); stable for x -> +/- inf.
    // v_exp_f32 + v_add_f32 + v_rcp_f32 + v_fma_f32 (no IEEE div sequence).
    float e = __expf(2.0f * v);
    float r = __builtin_amdgcn_rcpf(e + 1.0f);
    return fmaf(-2.0f, r, 1.0f);
}

struct MlpCtx {
    v2f   a[5];            // A fragments: W1^T duplicated rows 0-5 / 8-13
    v8f   cinit;           // bias broadcast {b1[0..5],0,0}
    float w2[HID][OS];
    float b2v[OS];
    int   hi;              // lane >= 16
};

// x[20] is feature-major: x[d*5 + i] = seq[i][d]
__device__ __forceinline__ void mlp_step(const MlpCtx& ctx, const float x[20],
                                         float& p0, float& p1) {
    v8f acc = ctx.cinit;
#pragma unroll
    for (int c = 0; c < 5; ++c) {
        // B-fragment: vgpr v, lane l holds B[k = 4c + 2*(l>=16) + v][n = l%16],
        // and lane l privately owns batch (l%16)'s features.
        v2f bm;
        bm.x = ctx.hi ? x[4 * c + 2] : x[4 * c + 0];
        bm.y = ctx.hi ? x[4 * c + 3] : x[4 * c + 1];
        acc = __builtin_amdgcn_wmma_f32_16x16x4_f32(
            /*neg_a=*/false, ctx.a[c], /*neg_b=*/false, bm,
            /*c_mod=*/(short)0, acc, /*reuse_a=*/false, /*reuse_b=*/false);
    }
    // Every lane now holds h0..h5 of its batch in acc[0..5].
    p0 = ctx.b2v[0];
    p1 = ctx.b2v[1];
#pragma unroll
    for (int j = 0; j < HID; ++j) {
        float h = tanh_fast(acc[j]);
        p0 = fmaf(h, ctx.w2[j][0], p0);
        p1 = fmaf(h, ctx.w2[j][1], p1);
    }
}

__global__ void __launch_bounds__(256)
or_mlp_rollout(const float* __restrict__ traj, const float* __restrict__ W1,
               const float* __restrict__ b1, const float* __restrict__ W2,
               const float* __restrict__ b2, float* __restrict__ out, int B) {
    const int lane = threadIdx.x & 31;
    const int wave = threadIdx.x >> 5;
    const int hi   = lane >> 4;
    const int n    = lane & 15;
    const int tileBase = (blockIdx.x * (int)(blockDim.x >> 5) + wave) * 16;
    if (tileBase >= B) return;          // wave-uniform: EXEC stays all-ones
    const int b = tileBase + n;         // lanes l and l+16 mirror batch l%16

    MlpCtx ctx;
    ctx.hi = hi;
    // A fragment: A[m][k] = (m in 0..5 or 8..13) ? W1[k][m%8] : 0
    // A layout: lane holds M = lane%16, K = 4c + 2*hi + vgpr.
    {
        const bool mv  = (n < HID) || (n >= 8 && n < 8 + HID);
        const int  col = (n < 8) ? n : (n - 8);
#pragma unroll
        for (int c = 0; c < 5; ++c) {
            float a0 = 0.0f, a1 = 0.0f;
            if (mv) {
                const int k0 = 4 * c + 2 * hi;
                a0 = W1[k0 * HID + col];
                a1 = W1[(k0 + 1) * HID + col];
            }
            ctx.a[c].x = a0;
            ctx.a[c].y = a1;
        }
    }
    // C init: row M gets b1[M%8] for valid rows -> same per-vgpr constant.
    ctx.cinit = v8f{b1[0], b1[1], b1[2], b1[3], b1[4], b1[5], 0.0f, 0.0f};
#pragma unroll
    for (int j = 0; j < HID; ++j) {
        ctx.w2[j][0] = W2[j * OS + 0];
        ctx.w2[j][1] = W2[j * OS + 1];
    }
    ctx.b2v[0] = b2[0];
    ctx.b2v[1] = b2[1];

    float2 o[WS];

    // ---- init step 0: seq = traj[:, 0:5, :] ----
    {
        float x[20];
        float4 rlast;
#pragma unroll
        for (int i = 0; i < WS; ++i) {
            float4 r = ld_row(traj, b, i);
            x[i] = r.x; x[5 + i] = r.y; x[10 + i] = r.z; x[15 + i] = r.w;
            if (i == WS - 1) rlast = r;
        }
        float p0, p1;
        mlp_step(ctx, x, p0, p1);
        o[0].x = rlast.z + p0;      // traj[:,4,2:4] + pred
        o[0].y = rlast.w + p1;
    }

    // ---- init steps t = 1..4 ----
#pragma unroll
    for (int t = 1; t < WS; ++t) {
        float x[20];
#pragma unroll
        for (int i = 0; i < WS; ++i) {
            if (i < WS - t) {
                float4 r = ld_row(traj, b, t + i);           // full traj row
                x[i] = r.x; x[5 + i] = r.y; x[10 + i] = r.z; x[15 + i] = r.w;
            } else {
                const int j = i - (WS - t);
                float4 r = ld_row(traj, b, WS + t - 1 + j);  // ctrl part only
                x[i] = r.x; x[5 + i] = r.y; x[10 + i] = o[j].x; x[15 + i] = o[j].y;
            }
        }
        float p0, p1;
        mlp_step(ctx, x, p0, p1);
        o[t].x = o[t - 1].x + p0;
        o[t].y = o[t - 1].y + p1;
    }

    // store the 5 warm-up outputs (lower half-wave only; upper is a mirror)
    if (!hi) {
#pragma unroll
        for (int t = 0; t < WS; ++t)
            *reinterpret_cast<float2*>(out + ((size_t)b * T_OUT + t) * OS) = o[t];
    }

    // ---- steady-state scan: t = 5 .. 250 ----
    float2 cw[WS];  // ctrl window traj[:, t..t+4, 0:2]
#pragma unroll
    for (int i = 0; i < WS; ++i) {
        float4 r = ld_row(traj, b, WS + i);
        cw[i].x = r.x; cw[i].y = r.y;
    }
    float2 w[WS];   // prediction window
#pragma unroll
    for (int i = 0; i < WS; ++i) w[i] = o[i];

    for (int t = WS; t < T_OUT; ++t) {
        float x[20];
#pragma unroll
        for (int i = 0; i < WS; ++i) {
            x[i] = cw[i].x; x[5 + i] = cw[i].y;
            x[10 + i] = w[i].x; x[15 + i] = w[i].y;
        }
        float p0, p1;
        mlp_step(ctx, x, p0, p1);
        float2 nw;
        nw.x = w[WS - 1].x + p0;
        nw.y = w[WS - 1].y + p1;
#pragma unroll
        for (int i = 0; i < WS - 1; ++i) { w[i] = w[i + 1]; cw[i] = cw[i + 1]; }
        w[WS - 1] = nw;
        {
            float4 r = ld_row(traj, b, t + WS);   // max index 255 < T_IN
            cw[WS - 1].x = r.x; cw[WS - 1].y = r.y;
        }
        if (!hi)
            *reinterpret_cast<float2*>(out + ((size_t)b * T_OUT + t) * OS) = nw;
    }
}

extern "C" void kernel_launch(void* const* d_in, const int* in_sizes, int n_in,
                              void* d_out, int out_size, void* d_ws, size_t ws_size,
                              hipStream_t stream) {
    const float* traj = (const float*)d_in[0];
    const float* W1   = (const float*)d_in[1];
    const float* b1   = (const float*)d_in[2];
    const float* W2   = (const float*)d_in[3];
    const float* b2   = (const float*)d_in[4];
    float* out = (float*)d_out;

    const int B = in_sizes[0] / (T_IN * DT);          // 32768
    const int block = 256;                            // 8 waves
    const int batchPerBlock = (block / 32) * 16;      // 128 batch elements
    const int grid = (B + batchPerBlock - 1) / batchPerBlock;
    or_mlp_rollout<<<grid, block, 0, stream>>>(traj, W1, b1, W2, b2, out, B);
}